// BACKBONE_47459388620824
// MI455X (gfx1250) — compile-verified
//
#include <hip/hip_runtime.h>

typedef float v2f __attribute__((ext_vector_type(2)));
typedef float v8f __attribute__((ext_vector_type(8)));

#define SW_ 240.0f
#define SH_ 180.0f
#define EPS_ 1e-5f

static __device__ __forceinline__ float fixf(float v) {
  // non-finite (NaN/±inf) -> 0 ; NaN fails the compare too
  return (fabsf(v) <= 3.402823466e38f) ? v : 0.0f;
}

// Monotonic float->int key: preserves total order, so a single signed-int
// atomic max implements float max with no divergence. Branchless.
static __device__ __forceinline__ int fkey(float v) {
  int f = __float_as_int(v);
  return (f < 0) ? (f ^ 0x7fffffff) : f;
}
// Inverse transform (key stored as float bits in the agg buffer).
static __device__ __forceinline__ float fdecode(float raw) {
  int k = __float_as_int(raw);
  int f = (k < 0) ? (k ^ 0x7fffffff) : k;
  return __int_as_float(f);
}
// Key-space init: INT_MIN == bits of -0.0f; decodes to NaN -> fixf -> 0,
// matching jnp.where(isfinite(agg), agg, 0) for empty segments.
#define KEY_INIT_F (-0.0f)

static __device__ __forceinline__ void atomicMaxKey(float* addr, float val) {
  atomicMax((int*)addr, fkey(val));
}

__global__ void k_fill(float* __restrict__ p, float v, int n) {
  int i = blockIdx.x * blockDim.x + threadIdx.x;
  if (i < n) p[i] = v;
}

// ---------------- block-1 edge convs (fp32 WMMA 16x16x4) ----------------
// Wave handles 16 edges. A(16x4): lanes 0-15 hold row M=l features K=0,1;
// lanes 16-31 hold K=2,3. B(4x16): VGPR0 = rows K=0 (lanes<16)/K=2 (lanes>=16),
// VGPR1 = rows K=1 / K=3, col N = lane&15. D: VGPR v, lane<16 -> edge v,
// lane>=16 -> edge v+8, channel = lane&15.

__global__ void k_edge_conv1(const float* __restrict__ x, const float* __restrict__ pos,
                             const int* __restrict__ src, const int* __restrict__ dst,
                             const float* __restrict__ W /*3x16*/, const float* __restrict__ B /*16*/,
                             float* __restrict__ agg /*Nx16 keys, pre KEY_INIT*/, int E) {
  int tid  = blockIdx.x * blockDim.x + threadIdx.x;
  int lane = threadIdx.x & 31;
  int wave = tid >> 5;
  int e0   = wave * 16;
  if (e0 >= E) return;
  int half = lane >> 4;
  int l    = lane & 15;
  if (e0 + 16 <= E) {                       // wave-uniform: EXEC stays full
    if (e0 + 4112 <= E) __builtin_prefetch(&src[e0 + 4096], 0, 0);
    int e = e0 + l;                          // lane j and j+16 hold edge e0+j
    int s = src[e], d = dst[e];
    float dx = pos[3 * s]     - pos[3 * d];
    float dy = pos[3 * s + 1] - pos[3 * d + 1];
    float xs = x[s];
    v2f a, b;
    a.x = half ? dy   : xs;
    a.y = half ? 0.0f : dx;
    b.x = half ? W[32 + l] : W[l];
    b.y = half ? 0.0f      : W[16 + l];
    v8f c;
    float bias = B[l];
    #pragma unroll
    for (int v = 0; v < 8; ++v) c[v] = bias;
    v8f dm = __builtin_amdgcn_wmma_f32_16x16x4_f32(false, a, false, b, (short)0, c, false, false);
    #pragma unroll
    for (int v = 0; v < 8; ++v) {
      int dd = __shfl(d, v + 8 * half, 32);  // dst of edge e0+v(+8) via ds_bpermute
      atomicMaxKey(&agg[dd * 16 + l], dm[v]);
    }
  } else {                                   // scalar tail (<16 edges)
    int e = e0 + lane;
    if (lane < 16 && e < E) {
      int s = src[e], d = dst[e];
      float xs = x[s];
      float dx = pos[3 * s]     - pos[3 * d];
      float dy = pos[3 * s + 1] - pos[3 * d + 1];
      for (int j = 0; j < 16; ++j) {
        float m = xs * W[j] + dx * W[16 + j] + dy * W[32 + j] + B[j];
        atomicMaxKey(&agg[d * 16 + j], m);
      }
    }
  }
}

__global__ void k_edge_conv2(const float* __restrict__ h /*Nx16*/, const float* __restrict__ pos,
                             const int* __restrict__ src, const int* __restrict__ dst,
                             const float* __restrict__ W /*18x16*/, const float* __restrict__ B /*16*/,
                             float* __restrict__ agg, int E) {
  int tid  = blockIdx.x * blockDim.x + threadIdx.x;
  int lane = threadIdx.x & 31;
  int wave = tid >> 5;
  int e0   = wave * 16;
  if (e0 >= E) return;
  int half = lane >> 4;
  int l    = lane & 15;
  if (e0 + 16 <= E) {
    if (e0 + 4112 <= E) __builtin_prefetch(&src[e0 + 4096], 0, 0);
    int e = e0 + l;
    int s = src[e], d = dst[e];
    float dx = pos[3 * s]     - pos[3 * d];
    float dy = pos[3 * s + 1] - pos[3 * d + 1];
    const float2* hv = (const float2*)(h + (long)s * 16);   // 8B-aligned pairs
    v8f acc;
    float bias = B[l];
    #pragma unroll
    for (int v = 0; v < 8; ++v) acc[v] = bias;
    #pragma unroll
    for (int kk = 0; kk < 5; ++kk) {         // K padded 18 -> 20
      int k0 = 4 * kk + 2 * half;
      v2f a, b;
      if (kk < 4) { float2 f = hv[k0 >> 1]; a.x = f.x; a.y = f.y; }
      else        { a.x = half ? 0.0f : dx; a.y = half ? 0.0f : dy; }
      int r0 = (k0     < 17) ? k0     : 17;
      int r1 = (k0 + 1 < 17) ? k0 + 1 : 17;
      float w0 = W[r0 * 16 + l], w1 = W[r1 * 16 + l];
      b.x = (k0     < 18) ? w0 : 0.0f;
      b.y = (k0 + 1 < 18) ? w1 : 0.0f;
      acc = __builtin_amdgcn_wmma_f32_16x16x4_f32(false, a, false, b, (short)0, acc, false, false);
    }
    #pragma unroll
    for (int v = 0; v < 8; ++v) {
      int dd = __shfl(d, v + 8 * half, 32);
      atomicMaxKey(&agg[dd * 16 + l], acc[v]);
    }
  } else {
    int e = e0 + lane;
    if (lane < 16 && e < E) {
      int s = src[e], d = dst[e];
      float dx = pos[3 * s]     - pos[3 * d];
      float dy = pos[3 * s + 1] - pos[3 * d + 1];
      const float* hs = h + (long)s * 16;
      for (int j = 0; j < 16; ++j) {
        float m = B[j] + dx * W[256 + j] + dy * W[272 + j];
        for (int ic = 0; ic < 16; ++ic) m += hs[ic] * W[ic * 16 + j];
        atomicMaxKey(&agg[d * 16 + j], m);
      }
    }
  }
}

// ---------------- batch-norm (column mean/var), with key-decode + fixup ----------------
__global__ void k_stats(const float* __restrict__ in, int N, int C,
                        float* __restrict__ stats, int dec) {
  __shared__ float ssum[128], ssq[128];
  int t = threadIdx.x;
  if (t < C) { ssum[t] = 0.0f; ssq[t] = 0.0f; }
  __syncthreads();
  int tc  = t % C;
  int rpb = blockDim.x / C;
  int r0  = blockIdx.x * rpb + t / C;
  int rst = gridDim.x * rpb;
  float s = 0.0f, q = 0.0f;
  for (int r = r0; r < N; r += rst) {
    float v = in[(long)r * C + tc];
    if (dec) v = fdecode(v);
    v = fixf(v);
    s += v; q += v * v;
  }
  atomicAdd(&ssum[tc], s);
  atomicAdd(&ssq[tc],  q);
  __syncthreads();
  if (t < C) { atomicAdd(&stats[t], ssum[t]); atomicAdd(&stats[C + t], ssq[t]); }
}

__global__ void k_bn(const float* __restrict__ in, float* __restrict__ out,
                     const float* __restrict__ stats, int N, int C, int relu, int dec) {
  int total = N * C;
  for (int i = blockIdx.x * blockDim.x + threadIdx.x; i < total; i += gridDim.x * blockDim.x) {
    int c = i % C;
    float mean = stats[c] / (float)N;
    float var  = stats[C + c] / (float)N - mean * mean;
    float v = in[i];
    if (dec) v = fdecode(v);
    v = (fixf(v) - mean) * rsqrtf(var + EPS_);
    if (relu) v = fmaxf(v, 0.0f);
    out[i] = v;
  }
}

__global__ void k_add_relu(const float* __restrict__ a, const float* __restrict__ b,
                           float* __restrict__ out, int n) {
  int i = blockIdx.x * blockDim.x + threadIdx.x;
  if (i < n) out[i] = fmaxf(a[i] + b[i], 0.0f);
}

// skip for block1: x (Nx1) @ lw(1x16) + lb
__global__ void k_skip1(const float* __restrict__ x, const float* __restrict__ lw,
                        const float* __restrict__ lb, float* __restrict__ out, int N) {
  int idx = blockIdx.x * blockDim.x + threadIdx.x;
  if (idx >= N * 16) return;
  int i = idx >> 4, c = idx & 15;
  out[idx] = x[i] * lw[c] + lb[c];
}

// generic dense: out(M x Cout) = x(M x Cin) @ W(Cin x Cout) + b
__global__ void k_lin(const float* __restrict__ x, const float* __restrict__ W,
                      const float* __restrict__ B, float* __restrict__ out,
                      int M, int Cin, int Cout) {
  int idx = blockIdx.x * blockDim.x + threadIdx.x;
  if (idx >= M * Cout) return;
  int i = idx / Cout, oc = idx % Cout;
  const float* row = x + (long)i * Cin;
  float acc = B[oc];
  for (int ic = 0; ic < Cin; ++ic) acc += row[ic] * W[ic * Cout + oc];
  out[idx] = acc;
}

// ---------------- pools ----------------
__global__ void k_pool1_scatter(const float* __restrict__ hin, const float* __restrict__ pos,
                                float* __restrict__ pooled /*4800x16, zeroed*/, int N) {
  int idx = blockIdx.x * blockDim.x + threadIdx.x;
  if (idx >= N * 16) return;
  int i = idx >> 4, c = idx & 15;
  int cx = (int)(pos[3 * i]     * (80.0f / SW_));
  int cy = (int)(pos[3 * i + 1] * (60.0f / SH_));
  cx = cx < 0 ? 0 : (cx > 79 ? 79 : cx);
  cy = cy < 0 ? 0 : (cy > 59 ? 59 : cy);
  atomicAdd(&pooled[(cy * 80 + cx) * 16 + c], hin[idx]);
}

__global__ void k_pool_grid_sum(const float* __restrict__ fin, int gwf, int ghf, int C,
                                float* __restrict__ pooled) {
  int gwc = gwf >> 1, ghc = ghf >> 1;
  int idx = blockIdx.x * blockDim.x + threadIdx.x;
  if (idx >= gwc * ghc * C) return;
  int c = idx % C, cell = idx / C;
  int cx = cell % gwc, cy = cell / gwc;
  float s = 0.0f;
  for (int oy = 0; oy < 2; ++oy)
    for (int ox = 0; ox < 2; ++ox)
      s += fin[((cy * 2 + oy) * gwf + (cx * 2 + ox)) * C + c];
  pooled[idx] = s;
}

__global__ void k_pool_mlp(const float* __restrict__ pooled, const float* __restrict__ W,
                           const float* __restrict__ B, float* __restrict__ out, int M, int C) {
  int idx = blockIdx.x * blockDim.x + threadIdx.x;
  if (idx >= M * C) return;
  int cell = idx / C, oc = idx % C;
  const float* row = pooled + (long)cell * C;
  float v = B[oc];
  for (int ic = 0; ic < C; ++ic) v += row[ic] * W[ic * C + oc];
  out[idx] = (v > 1.0f) ? v : 0.0f;
}

// ---------------- grid conv: segment-max over <=9 analytic neighbors (incl. self) ----------------
__global__ void k_grid_conv(const float* __restrict__ hin, const float* __restrict__ W,
                            const float* __restrict__ B, float* __restrict__ out,
                            int gw, int gh, int Cin, int Cout) {
  int M = gw * gh;
  int idx = blockIdx.x * blockDim.x + threadIdx.x;
  if (idx >= M * Cout) return;
  int node = idx / Cout, oc = idx % Cout;
  int cx = node % gw, cy = node / gw;
  float sw = SW_ / (float)gw, sh = SH_ / (float)gh;
  float wdx = W[Cin * Cout + oc], wdy = W[(Cin + 1) * Cout + oc];
  float best = -3.402823466e38f;
  for (int oy = -1; oy <= 1; ++oy) {
    for (int ox = -1; ox <= 1; ++ox) {
      int sx = cx + ox, sy = cy + oy;
      if (sx < 0 || sx >= gw || sy < 0 || sy >= gh) continue;
      const float* hs = hin + (long)(sy * gw + sx) * Cin;
      float acc = B[oc] + (float)ox * sw * wdx + (float)oy * sh * wdy;
      for (int ic = 0; ic < Cin; ++ic) acc += hs[ic] * W[ic * Cout + oc];
      best = fmaxf(best, acc);
    }
  }
  out[idx] = best;   // self-edge guarantees finite
}

// ---------------- host orchestration ----------------
static inline int cdiv_i(int a, int b) { return (a + b - 1) / b; }

static void run_bn(const float* in, float* out, float* stats, int N, int C, int relu, int dec,
                   hipStream_t stream) {
  k_fill<<<1, 256, 0, stream>>>(stats, 0.0f, 2 * C);
  int rpb = 256 / C;
  int blocks = cdiv_i(N, rpb); if (blocks > 2048) blocks = 2048;
  k_stats<<<blocks, 256, 0, stream>>>(in, N, C, stats, dec);
  k_bn<<<cdiv_i(N * C, 256), 256, 0, stream>>>(in, out, stats, N, C, relu, dec);
}

static void run_grid_block(const float* xin, int gw, int gh, int Cin, int Cout,
                           const float* c1w, const float* c1b, const float* c2w, const float* c2b,
                           const float* lw, const float* lb,
                           float* t0, float* t1, float* t2, float* stats,
                           float* outFinal, hipStream_t stream) {
  int M = gw * gh;
  k_grid_conv<<<cdiv_i(M * Cout, 256), 256, 0, stream>>>(xin, c1w, c1b, t0, gw, gh, Cin, Cout);
  run_bn(t0, t0, stats, M, Cout, 1, 0, stream);
  k_grid_conv<<<cdiv_i(M * Cout, 256), 256, 0, stream>>>(t0, c2w, c2b, t1, gw, gh, Cout, Cout);
  run_bn(t1, t1, stats, M, Cout, 0, 0, stream);
  k_lin<<<cdiv_i(M * Cout, 256), 256, 0, stream>>>(xin, lw, lb, t2, M, Cin, Cout);
  run_bn(t2, t2, stats, M, Cout, 0, 0, stream);
  k_add_relu<<<cdiv_i(M * Cout, 256), 256, 0, stream>>>(t1, t2, outFinal, M * Cout);
}

extern "C" void kernel_launch(void* const* d_in, const int* in_sizes, int n_in,
                              void* d_out, int out_size, void* d_ws, size_t ws_size,
                              hipStream_t stream) {
  const float* x   = (const float*)d_in[0];
  const float* pos = (const float*)d_in[1];
  const int*   ei  = (const int*)d_in[2];
  int N = in_sizes[0];
  int E = in_sizes[2] / 2;
  const int* src = ei;
  const int* dst = ei + E;
  const float* P[30];
  for (int i = 0; i < 30; ++i) P[i] = (const float*)d_in[3 + i];
  // P[0..5]=b1{c1w,c1b,c2w,c2b,lw,lb}, P[6..11]=b2, P[12..17]=b3, P[18..23]=b4,
  // P[24..29]=p1w,p1b,p2w,p2b,p3w,p3b
  float* fout = (float*)d_out;

  float* ws    = (float*)d_ws;
  size_t o = 0;
  float* bufA  = ws + o; o += (size_t)N * 16;
  float* bufB  = ws + o; o += (size_t)N * 16;
  float* bufC  = ws + o; o += (size_t)N * 16;
  float* stats = ws + o; o += 256;
  float* pooled1 = ws + o; o += 4800 * 16;
  float* x1      = ws + o; o += 4800 * 16;
  float* ga      = ws + o; o += 4800 * 32;
  float* gb      = ws + o; o += 4800 * 32;
  float* gc      = ws + o; o += 4800 * 32;
  float* pooled2 = ws + o; o += 1200 * 32;
  float* x2      = ws + o; o += 1200 * 32;
  float* pooled3 = ws + o; o += 300 * 64;
  float* x3      = ws + o; o += 300 * 64;
  (void)ws_size; (void)n_in; (void)out_size;

  int waves  = cdiv_i(E, 16);
  int eblks  = cdiv_i(waves * 32, 256);

  // ---- block 1 (edge graph) ----
  k_skip1<<<cdiv_i(N * 16, 256), 256, 0, stream>>>(x, P[4], P[5], bufC, N);
  run_bn(bufC, bufC, stats, N, 16, 0, 0, stream);

  k_fill<<<cdiv_i(N * 16, 256), 256, 0, stream>>>(bufA, KEY_INIT_F, N * 16);
  k_edge_conv1<<<eblks, 256, 0, stream>>>(x, pos, src, dst, P[0], P[1], bufA, E);
  run_bn(bufA, bufB, stats, N, 16, 1, 1, stream);        // h = relu(bn(decode(conv1)))

  k_fill<<<cdiv_i(N * 16, 256), 256, 0, stream>>>(bufA, KEY_INIT_F, N * 16);
  k_edge_conv2<<<eblks, 256, 0, stream>>>(bufB, pos, src, dst, P[2], P[3], bufA, E);
  run_bn(bufA, bufA, stats, N, 16, 0, 1, stream);        // h2 = bn(decode(conv2))

  k_add_relu<<<cdiv_i(N * 16, 256), 256, 0, stream>>>(bufA, bufC, bufB, N * 16); // h1 final

  // ---- pool 1 -> 80x60 grid ----
  k_fill<<<cdiv_i(4800 * 16, 256), 256, 0, stream>>>(pooled1, 0.0f, 4800 * 16);
  k_pool1_scatter<<<cdiv_i(N * 16, 256), 256, 0, stream>>>(bufB, pos, pooled1, N);
  k_pool_mlp<<<cdiv_i(4800 * 16, 256), 256, 0, stream>>>(pooled1, P[24], P[25], x1, 4800, 16);

  // ---- block 2 -> feat0 (d_out + 0) ----
  run_grid_block(x1, 80, 60, 16, 32, P[6], P[7], P[8], P[9], P[10], P[11],
                 ga, gb, gc, stats, fout, stream);

  // ---- pool 2 -> 40x30 ----
  k_pool_grid_sum<<<cdiv_i(1200 * 32, 256), 256, 0, stream>>>(fout, 80, 60, 32, pooled2);
  k_pool_mlp<<<cdiv_i(1200 * 32, 256), 256, 0, stream>>>(pooled2, P[26], P[27], x2, 1200, 32);

  // ---- block 3 -> feat1 (d_out + 153600) ----
  run_grid_block(x2, 40, 30, 32, 64, P[12], P[13], P[14], P[15], P[16], P[17],
                 ga, gb, gc, stats, fout + 153600, stream);

  // ---- pool 3 -> 20x15 ----
  k_pool_grid_sum<<<cdiv_i(300 * 64, 256), 256, 0, stream>>>(fout + 153600, 40, 30, 64, pooled3);
  k_pool_mlp<<<cdiv_i(300 * 64, 256), 256, 0, stream>>>(pooled3, P[28], P[29], x3, 300, 64);

  // ---- block 4 -> feat2 (d_out + 230400) ----
  run_grid_block(x3, 20, 15, 64, 128, P[18], P[19], P[20], P[21], P[22], P[23],
                 ga, gb, gc, stats, fout + 230400, stream);
}